// DyfConv_11424613007826
// MI455X (gfx1250) — compile-verified
//
#include <hip/hip_runtime.h>
#include <hip/hip_bf16.h>
#include <math.h>

typedef __attribute__((ext_vector_type(2))) float v2f;
typedef __attribute__((ext_vector_type(8))) float v8f;

// problem constants (fixed by the reference setup)
constexpr int KSZ   = 3;
constexpr int K2    = 9;            // 3*3 taps
constexpr int CIN   = 64;
constexpr int COUT  = 64;
constexpr int NB    = 8;            // batch
constexpr int H     = 128;
constexpr int W     = 128;
constexpr int KDIM  = CIN * K2;     // 576 = GEMM reduction dim

constexpr int THREADS     = 256;    // 8 wave32
constexpr int PIX_PER_BLK = 32;     // 2 M-tiles of 16 pixels
constexpr int YSTRIDE     = KDIM + 1; // 577: odd stride -> conflict-free A loads

// LDS partition (floats):
//   ldsW : KDIM*COUT        = 36864   (staged weight, B-matrix)
//   ldsY : 32*YSTRIDE       = 18464   (per-pixel operated patches, A-matrix)
//   ldsOp: 32*81            =  2592   (per-pixel dynamic 9x9 operator)
//   ldsN : 32*9             =   288   (per-pixel sum-of-squares accumulators)
constexpr int LDS_W_OFF  = 0;
constexpr int LDS_Y_OFF  = LDS_W_OFF + KDIM * COUT;
constexpr int LDS_OP_OFF = LDS_Y_OFF + PIX_PER_BLK * YSTRIDE;
constexpr int LDS_N_OFF  = LDS_OP_OFF + PIX_PER_BLK * 81;
constexpr int LDS_FLOATS = LDS_N_OFF + PIX_PER_BLK * K2;

__global__ __launch_bounds__(THREADS, 1)
void dyfconv_fused(const float* __restrict__ x,
                   const float* __restrict__ weight,   // (CIN, COUT, 3, 3)
                   const float* __restrict__ bias,     // (COUT)
                   const float* __restrict__ op_w,     // (81, 9)
                   const float* __restrict__ op_b,     // (81)
                   float* __restrict__ out)            // (NB, COUT, H, W)
{
    extern __shared__ float lds[];
    float* ldsW  = lds + LDS_W_OFF;
    float* ldsY  = lds + LDS_Y_OFF;
    float* ldsOp = lds + LDS_OP_OFF;
    float* ldsN  = lds + LDS_N_OFF;

    const int t     = threadIdx.x;
    const int pbase = blockIdx.x * PIX_PER_BLK;

    // ---- stage GEMM B-matrix: ldsW[kk*COUT + o] = weight[c][o][k], kk = c*9+k
    for (int idx = t; idx < KDIM * COUT; idx += THREADS) {
        int kk = idx >> 6;          // / COUT
        int o  = idx & (COUT - 1);
        int c  = kk / K2;
        int k  = kk - c * K2;
        ldsW[idx] = weight[(c * COUT + o) * K2 + k];
    }
    for (int idx = t; idx < PIX_PER_BLK * K2; idx += THREADS)
        ldsN[idx] = 0.0f;
    __syncthreads();

    // ---- per-thread mapping for phase 1: pixel p, channel group g (8 ch each)
    const int p  = t >> 3;          // 0..31
    const int g  = t & 7;           // 0..7
    const int pg = pbase + p;       // global pixel
    const int bb = pg >> 14;        // / (H*W)
    const int hw = pg & (H * W - 1);
    const int hh = hw >> 7;
    const int ww = hw & (W - 1);

    // ---- phase 1a: partial sum of squares over this thread's 8 channels
    {
        float sq[K2];
        #pragma unroll
        for (int j = 0; j < K2; ++j) sq[j] = 0.0f;
        for (int cc = 0; cc < 8; ++cc) {
            const int c = g * 8 + cc;
            const float* xc = x + (size_t)(bb * CIN + c) * (H * W);
            #pragma unroll
            for (int ki = 0; ki < KSZ; ++ki) {
                const int hs  = hh + ki - 1;
                const bool okh = (unsigned)hs < (unsigned)H;
                #pragma unroll
                for (int kj = 0; kj < KSZ; ++kj) {
                    const int wsv = ww + kj - 1;
                    const bool ok = okh && ((unsigned)wsv < (unsigned)W);
                    const float v = ok ? xc[hs * W + wsv] : 0.0f;
                    sq[ki * KSZ + kj] += v * v;
                }
            }
        }
        #pragma unroll
        for (int j = 0; j < K2; ++j)
            atomicAdd(&ldsN[p * K2 + j], sq[j]);   // ds_add_f32
    }
    __syncthreads();

    // ---- phase 1b: dynamic operator  op = tanh(n @ op_w^T + op_b)
    for (int e = t; e < PIX_PER_BLK * 81; e += THREADS) {
        const int pp  = e / 81;
        const int idx = e - pp * 81;
        const float* nb = &ldsN[pp * K2];
        float acc = op_b[idx];
        #pragma unroll
        for (int j = 0; j < K2; ++j)
            acc += sqrtf(nb[j]) * op_w[idx * K2 + j];
        ldsOp[e] = tanhf(acc);
    }
    __syncthreads();

    // ---- phase 1c: y[c][k] = sum_j xu[c][j] * (op[j][k] + (j==k))
    {
        const float* opp = &ldsOp[p * 81];
        float* yrow = &ldsY[p * YSTRIDE];
        for (int cc = 0; cc < 8; ++cc) {
            const int c = g * 8 + cc;
            const float* xc = x + (size_t)(bb * CIN + c) * (H * W);
            float xu[K2];
            #pragma unroll
            for (int ki = 0; ki < KSZ; ++ki) {
                const int hs  = hh + ki - 1;
                const bool okh = (unsigned)hs < (unsigned)H;
                #pragma unroll
                for (int kj = 0; kj < KSZ; ++kj) {
                    const int wsv = ww + kj - 1;
                    const bool ok = okh && ((unsigned)wsv < (unsigned)W);
                    xu[ki * KSZ + kj] = ok ? xc[hs * W + wsv] : 0.0f;
                }
            }
            #pragma unroll
            for (int k = 0; k < K2; ++k) {
                float acc = xu[k];                   // identity term (x_op + xu)
                #pragma unroll
                for (int j = 0; j < K2; ++j)
                    acc += xu[j] * opp[j * K2 + k];
                yrow[c * K2 + k] = acc;
            }
        }
    }
    __syncthreads();

    // ---- phase 2: WMMA f32 GEMM, one 16x16 (pixels x out-ch) tile per wave
    const int wave  = t >> 5;
    const int lane  = t & 31;
    const int mTile = wave & 1;        // 2 pixel tiles
    const int nTile = wave >> 1;       // 4 out-channel tiles
    const int mIdx  = lane & 15;       // A: M = lane%16
    const int nIdx  = lane & 15;       // B/C: N = lane%16
    const int kh    = lane >> 4;       // lane half selects K pair {0,1} vs {2,3}

    const float* yrow = &ldsY[(mTile * 16 + mIdx) * YSTRIDE];
    const float* wcol = &ldsW[nTile * 16 + nIdx];

    v8f acc = {};
    #pragma unroll 4
    for (int kk = 0; kk < KDIM; kk += 4) {
        v2f a, bf;
        const int kb = kk + 2 * kh;
        // A 16x4 f32: v0 = A[M][2*kh], v1 = A[M][2*kh+1]
        a.x = yrow[kb + 0];
        a.y = yrow[kb + 1];
        // B 4x16 f32: v0 = B[2*kh][N], v1 = B[2*kh+1][N]
        bf.x = wcol[(kb + 0) * COUT];
        bf.y = wcol[(kb + 1) * COUT];
        acc = __builtin_amdgcn_wmma_f32_16x16x4_f32(
                  /*neg_a=*/false, a, /*neg_b=*/false, bf,
                  /*c_mod=*/(short)0, acc, /*reuse_a=*/false, /*reuse_b=*/false);
    }

    // ---- epilogue: C/D layout -> acc[r] = C[M = r + 8*kh][N = nIdx]
    const int o  = nTile * 16 + nIdx;
    const float bo = bias[o];
    #pragma unroll
    for (int r = 0; r < 8; ++r) {
        const int m     = r + 8 * kh;
        const int pglob = pbase + mTile * 16 + m;
        const int b_    = pglob >> 14;
        const int hw_   = pglob & (H * W - 1);
        out[((size_t)(b_ * COUT + o)) * (H * W) + hw_] = acc[r] + bo;
    }
}

extern "C" void kernel_launch(void* const* d_in, const int* in_sizes, int n_in,
                              void* d_out, int out_size, void* d_ws, size_t ws_size,
                              hipStream_t stream) {
    (void)in_sizes; (void)n_in; (void)out_size; (void)d_ws; (void)ws_size;
    const float* x      = (const float*)d_in[0];
    const float* weight = (const float*)d_in[1];
    const float* bias   = (const float*)d_in[2];
    const float* op_w   = (const float*)d_in[3];
    const float* op_b   = (const float*)d_in[4];
    float* out = (float*)d_out;

    const int totalPix = NB * H * W;                 // 131072
    const int blocks   = totalPix / PIX_PER_BLK;     // 4096
    const size_t ldsBytes = (size_t)LDS_FLOATS * sizeof(float); // ~227 KB (<320 KB WGP LDS)

    dyfconv_fused<<<blocks, THREADS, ldsBytes, stream>>>(x, weight, bias, op_w, op_b, out);
}